// E3nnInteractionBlock_33509334843735
// MI455X (gfx1250) — compile-verified
//
#include <hip/hip_runtime.h>
#include <math.h>
#include <utility>

typedef float v2f __attribute__((ext_vector_type(2)));
typedef float v8f __attribute__((ext_vector_type(8)));

#define HIDDEN  32
#define N_NODES 4096
#define N_EDGES 8192
#define FEAT    512
#define NPATHS  23

// Path table in the reference's enumeration order (l1 outer, l2 mid, l3 inner).
static constexpr int PL1[NPATHS] = {0,0,0,0, 1,1,1,1,1,1, 2,2,2,2,2,2,2, 3,3,3,3,3,3};
static constexpr int PL2[NPATHS] = {0,1,2,3, 0,1,1,2,2,3, 0,1,1,2,2,3,3, 0,1,2,2,3,3};
static constexpr int PL3[NPATHS] = {0,1,2,3, 1,0,2,1,3,2, 2,1,3,0,2,1,3, 3,2,1,3,0,2};
// PATH_W[l3] = sqrt((2*l3+1)/(cnt[l3]*HIDDEN)), cnt = {4,6,7,6}  (precomputed)
static constexpr float PWT[4] = {
  0.08838834764831845f,   // sqrt(1/128)
  0.125f,                 // sqrt(3/192)
  0.14940357616680451f,   // sqrt(5/224)
  0.19094065395649326f    // sqrt(7/192)
};

// Real spherical harmonics (l=0..3), same normalization as reference _sh().
template <typename T>
__device__ __forceinline__ void sh_eval(T x, T y, T z, T* Y) {
  const T s3  = (T)1.7320508075688772;
  const T s5  = (T)2.2360679774997896;
  const T s7  = (T)2.6457513110645907;
  const T s15 = (T)3.8729833462074170;
  const T c58 = (T)0.7905694150420949;   // sqrt(5/8)
  const T c38 = (T)0.6123724356957945;   // sqrt(3/8)
  Y[0]  = (T)1;
  Y[1]  = s3 * x;
  Y[2]  = s3 * y;
  Y[3]  = s3 * z;
  Y[4]  = s5 * s3 * x * z;
  Y[5]  = s5 * s3 * x * y;
  Y[6]  = s5 * (y * y - (T)0.5 * (x * x + z * z));
  Y[7]  = s5 * s3 * y * z;
  Y[8]  = s5 * (s3 * (T)0.5) * (z * z - x * x);
  Y[9]  = s7 * c58 * x * ((T)3 * z * z - x * x);
  Y[10] = s7 * s15 * x * y * z;
  Y[11] = s7 * c38 * x * ((T)5 * y * y - (T)1);
  Y[12] = s7 * (T)0.5 * y * ((T)5 * y * y - (T)3);
  Y[13] = s7 * c38 * z * ((T)5 * y * y - (T)1);
  Y[14] = s7 * (s15 * (T)0.5) * y * (z * z - x * x);
  Y[15] = s7 * c58 * z * (z * z - (T)3 * x * x);
}

// ---------------------------------------------------------------------------
// Init kernel: normalized Wigner coupling tables C[p][i][j][k] (7x7x7 padded,
// fp32) via 16-pt Gauss-Legendre x 32-pt phi quadrature in double. 1 block/path.
// ---------------------------------------------------------------------------
__launch_bounds__(64)
__global__ void w3j_kernel(float* __restrict__ ctab) {
  const int p = blockIdx.x;
  const int t = threadIdx.x;

  int l1 = 0, l2 = 0, l3 = 0, cnt = 0;
  for (int a = 0; a < 4; ++a)
    for (int b = 0; b < 4; ++b)
      for (int c = 0; c < 4; ++c) {
        int lo = a - b; if (lo < 0) lo = -lo;
        if (lo <= c && c <= a + b && (((a + b + c) & 1) == 0)) {
          if (cnt == p) { l1 = a; l2 = b; l3 = c; }
          ++cnt;
        }
      }
  const int d1 = 2 * l1 + 1, d2 = 2 * l2 + 1, d3 = 2 * l3 + 1;
  const int NE = d1 * d2 * d3;

  double un[16], wn[16];
  const double PI = 3.14159265358979323846;
  for (int i = 0; i < 16; ++i) {
    double xg = cos(PI * (i + 0.75) / 16.5);
    double dp = 1.0;
    for (int it = 0; it < 100; ++it) {
      double a0 = 1.0, b0 = xg;
      for (int n = 2; n <= 16; ++n) {
        double c0 = ((2.0 * n - 1.0) * xg * b0 - (n - 1.0) * a0) / n;
        a0 = b0; b0 = c0;
      }
      dp = 16.0 * (xg * b0 - a0) / (xg * xg - 1.0);
      double dx = b0 / dp;
      xg -= dx;
      if (fabs(dx) < 1e-14) break;
    }
    {
      double a0 = 1.0, b0 = xg;
      for (int n = 2; n <= 16; ++n) {
        double c0 = ((2.0 * n - 1.0) * xg * b0 - (n - 1.0) * a0) / n;
        a0 = b0; b0 = c0;
      }
      dp = 16.0 * (xg * b0 - a0) / (xg * xg - 1.0);
    }
    un[i] = xg;
    wn[i] = 2.0 / ((1.0 - xg * xg) * dp * dp);
  }

  __shared__ double Cs[343];
  __shared__ double red[64];
  const double DPHI = 2.0 * PI / 32.0;

  for (int n = t; n < NE; n += 64) {
    const int i = n / (d2 * d3);
    const int j = (n / d3) % d2;
    const int k = n % d3;
    double s = 0.0;
    for (int q = 0; q < 512; ++q) {
      const int iq = q >> 5, jq = q & 31;
      const double uu  = un[iq];
      const double wq  = wn[iq] * DPHI;
      const double phi = jq * DPHI;
      double st2 = 1.0 - uu * uu;
      double st  = sqrt(st2 > 0.0 ? st2 : 0.0);
      double Yq[16];
      sh_eval(st * cos(phi), st * sin(phi), uu, Yq);
      s += wq * Yq[l1 * l1 + i] * Yq[l2 * l2 + j] * Yq[l3 * l3 + k];
    }
    Cs[n] = s;
  }
  __syncthreads();

  double ss = 0.0;
  for (int n = t; n < NE; n += 64) ss += Cs[n] * Cs[n];
  red[t] = ss;
  __syncthreads();
  for (int off = 32; off > 0; off >>= 1) {
    if (t < off) red[t] += red[t + off];
    __syncthreads();
  }
  const double inv = 1.0 / sqrt(red[0]);

  for (int n = t; n < NE; n += 64) {
    const int i = n / (d2 * d3);
    const int j = (n / d3) % d2;
    const int k = n % d3;
    ctab[p * 343 + (i * 7 + j) * 7 + k] = (float)(Cs[n] * inv);
  }
}

__global__ void zero_kernel(float* __restrict__ out, int n) {
  int i = blockIdx.x * blockDim.x + threadIdx.x;
  if (i < n) out[i] = 0.0f;
}

// ---------------------------------------------------------------------------
// Per-path body, fully specialized at compile time (template parameter P).
// One wave32 per edge. 16 x V_WMMA_F32_16X16X4_F32 per path.
// ---------------------------------------------------------------------------
template <int P>
__device__ __forceinline__ void path_body(
    int lane, int n16,
    float len,
    const float (&Y)[16], const float (&xr)[16],
    const float* __restrict__ wwp,   // w_weight + usel*32 + n16
    const float* __restrict__ wbp,   // w_bias   + usel*32 + n16
    const float* __restrict__ ctab,
    float* Gs, float* tmpB, const float* tb,  // tb = tmpB + usel*16 + n16
    v8f& accLo, v8f& accHi) {
  constexpr int l1 = PL1[P], l2 = PL2[P], l3 = PL3[P];
  constexpr int d1 = 2 * l1 + 1, d2 = 2 * l2 + 1, d3 = 2 * l3 + 1;
  constexpr int GN = d1 * d3;
  constexpr float pw = PWT[l3];
  const float* Cp = ctab + P * 343;

  // G[i][k] = PW * sum_j Y[j] * C[i][j][k]  (GN <= 49 -> two predicated blocks)
  if (lane < GN) {
    const int i = lane / d3, k = lane % d3;
    float g = 0.0f;
#pragma unroll
    for (int j = 0; j < d2; ++j) g = fmaf(Y[l2 * l2 + j], Cp[(i * 7 + j) * 7 + k], g);
    Gs[lane] = pw * g;
  }
  if constexpr (GN > 32) {
    const int n2 = lane + 32;
    if (n2 < GN) {
      const int i = n2 / d3, k = n2 % d3;
      float g = 0.0f;
#pragma unroll
      for (int j = 0; j < d2; ++j) g = fmaf(Y[l2 * l2 + j], Cp[(i * 7 + j) * 7 + k], g);
      Gs[n2] = pw * g;
    }
  }
  __syncthreads();

  // tmp[u=lane][k] = sum_i xr[i] * G[i][k]; write zero-padded B tile (32x16)
  float tk[7];
#pragma unroll
  for (int k = 0; k < d3; ++k) {
    float s = 0.0f;
#pragma unroll
    for (int i = 0; i < d1; ++i) s = fmaf(xr[l1 * l1 + i], Gs[i * d3 + k], s);
    tk[k] = s;
  }
#pragma unroll
  for (int k = 0; k < 16; ++k) tmpB[lane * 16 + k] = (k < d3) ? tk[k] : 0.0f;
  __syncthreads();

  // acc += wp^T x tmp : 8 K-steps of 4 over u, 2 w-halves.
  constexpr int wbase = P * (HIDDEN * HIDDEN);
#pragma unroll
  for (int kc = 0; kc < 8; ++kc) {
    v2f b;
    b.x = tb[kc * 64];
    b.y = tb[kc * 64 + 16];
    {
      const int i0 = wbase + kc * 128;       // h = 0
      v2f a;
      a.x = fmaf(len, wwp[i0], wbp[i0]);
      a.y = fmaf(len, wwp[i0 + HIDDEN], wbp[i0 + HIDDEN]);
      accLo = __builtin_amdgcn_wmma_f32_16x16x4_f32(
          false, a, false, b, (short)0, accLo, false, false);
    }
    {
      const int i0 = wbase + kc * 128 + 16;  // h = 1
      v2f a;
      a.x = fmaf(len, wwp[i0], wbp[i0]);
      a.y = fmaf(len, wwp[i0 + HIDDEN], wbp[i0 + HIDDEN]);
      accHi = __builtin_amdgcn_wmma_f32_16x16x4_f32(
          false, a, false, b, (short)0, accHi, false, false);
    }
  }
  __syncthreads();  // protect Gs/tmpB before next path rewrites them
}

template <int... Ps>
__device__ __forceinline__ void run_group(
    std::integer_sequence<int, Ps...>,
    int lane, int n16, float len,
    const float (&Y)[16], const float (&xr)[16],
    const float* __restrict__ wwp, const float* __restrict__ wbp,
    const float* __restrict__ ctab,
    float* Gs, float* tmpB, const float* tb,
    v8f& accLo, v8f& accHi) {
  (path_body<Ps>(lane, n16, len, Y, xr, wwp, wbp, ctab, Gs, tmpB, tb, accLo, accHi), ...);
}

// Scatter-add one l3 tile pair into out[dst]. C/D layout:
// VGPR r: lanes 0-15 -> M=r, lanes 16-31 -> M=r+8; N = lane&15.
template <int L3>
__device__ __forceinline__ void store_tile(int lane, int n16, float* od,
                                           const v8f& accLo, const v8f& accHi) {
  constexpr int d3 = 2 * L3 + 1;
  if (n16 < d3) {
    const int wadd = (lane >= 16) ? 8 : 0;
#pragma unroll
    for (int r = 0; r < 8; ++r) {
      const int w = r + wadd;
      atomicAdd(od + HIDDEN * L3 * L3 + w * d3 + n16, accLo[r]);
      atomicAdd(od + HIDDEN * L3 * L3 + (w + 16) * d3 + n16, accHi[r]);
    }
  }
}

__launch_bounds__(32)
__global__ void edge_kernel(const float* __restrict__ nf,
                            const int*   __restrict__ eidx,
                            const float* __restrict__ evec,
                            const float* __restrict__ ww,
                            const float* __restrict__ wb,
                            const float* __restrict__ ctab,
                            float*       __restrict__ out) {
  const int e    = blockIdx.x;
  const int lane = threadIdx.x;

  const int src = eidx[e];
  const int dst = eidx[N_EDGES + e];

  const float vx = evec[3 * e + 0];
  const float vy = evec[3 * e + 1];
  const float vz = evec[3 * e + 2];
  float len = sqrtf(vx * vx + vy * vy + vz * vz);
  len = fmaxf(len, 1e-8f);
  const float inv = 1.0f / len;

  float Y[16];
  sh_eval(vx * inv, vy * inv, vz * inv, Y);

  // Lane u holds row u of every l1 block: 1+3+5+7 = 16 floats (xr[l1*l1+i]).
  const float* xs = nf + (size_t)src * FEAT;
  float xr[16];
#pragma unroll
  for (int l = 0; l < 4; ++l) {
    const int d = 2 * l + 1;
#pragma unroll
    for (int i = 0; i < 7; ++i)
      if (i < d) xr[l * l + i] = xs[HIDDEN * l * l + lane * d + i];
  }

  __shared__ float Gs[49];
  __shared__ float tmpB[32 * 16];

  const int n16  = lane & 15;
  const int usel = (lane >> 4) << 1;   // 0 or 2: which K sub-pair this lane feeds
  const float* wwp = ww + usel * HIDDEN + n16;
  const float* wbp = wb + usel * HIDDEN + n16;
  const float* tb  = tmpB + usel * 16 + n16;

  float* od = out + (size_t)dst * FEAT;

  {  // l3 = 0 : paths (0,0,0) (1,1,0) (2,2,0) (3,3,0)
    v8f a0 = {0,0,0,0,0,0,0,0}, a1 = {0,0,0,0,0,0,0,0};
    run_group(std::integer_sequence<int, 0, 5, 13, 21>{},
              lane, n16, len, Y, xr, wwp, wbp, ctab, Gs, tmpB, tb, a0, a1);
    store_tile<0>(lane, n16, od, a0, a1);
  }
  {  // l3 = 1 : paths (0,1,1) (1,0,1) (1,2,1) (2,1,1) (2,3,1) (3,2,1)
    v8f a0 = {0,0,0,0,0,0,0,0}, a1 = {0,0,0,0,0,0,0,0};
    run_group(std::integer_sequence<int, 1, 4, 7, 11, 15, 19>{},
              lane, n16, len, Y, xr, wwp, wbp, ctab, Gs, tmpB, tb, a0, a1);
    store_tile<1>(lane, n16, od, a0, a1);
  }
  {  // l3 = 2 : paths (0,2,2) (1,1,2) (1,3,2) (2,0,2) (2,2,2) (3,1,2) (3,3,2)
    v8f a0 = {0,0,0,0,0,0,0,0}, a1 = {0,0,0,0,0,0,0,0};
    run_group(std::integer_sequence<int, 2, 6, 9, 10, 14, 18, 22>{},
              lane, n16, len, Y, xr, wwp, wbp, ctab, Gs, tmpB, tb, a0, a1);
    store_tile<2>(lane, n16, od, a0, a1);
  }
  {  // l3 = 3 : paths (0,3,3) (1,2,3) (2,1,3) (2,3,3) (3,0,3) (3,2,3)
    v8f a0 = {0,0,0,0,0,0,0,0}, a1 = {0,0,0,0,0,0,0,0};
    run_group(std::integer_sequence<int, 3, 8, 12, 16, 17, 20>{},
              lane, n16, len, Y, xr, wwp, wbp, ctab, Gs, tmpB, tb, a0, a1);
    store_tile<3>(lane, n16, od, a0, a1);
  }
}

extern "C" void kernel_launch(void* const* d_in, const int* in_sizes, int n_in,
                              void* d_out, int out_size, void* d_ws, size_t ws_size,
                              hipStream_t stream) {
  const float* nf   = (const float*)d_in[0]; // node_features (4096,512) f32
  const int*   eidx = (const int*)  d_in[1]; // edge_index (2,8192) i32
  const float* evec = (const float*)d_in[2]; // edge_vectors (8192,3) f32
  const float* ww   = (const float*)d_in[3]; // w_weight (23552,1) f32
  const float* wb   = (const float*)d_in[4]; // w_bias (23552,) f32
  float* out  = (float*)d_out;               // (4096,512) f32
  float* ctab = (float*)d_ws;                // 23*343 f32 coupling tables

  w3j_kernel<<<NPATHS, 64, 0, stream>>>(ctab);
  const int n = N_NODES * FEAT;
  zero_kernel<<<(n + 255) / 256, 256, 0, stream>>>(out, n);
  edge_kernel<<<N_EDGES, 32, 0, stream>>>(nf, eidx, evec, ww, wb, ctab, out);
}